// ScalableFFT_45801531245098
// MI455X (gfx1250) — compile-verified
//
#include <hip/hip_runtime.h>

// 2^24-point Walsh-Hadamard transform (reference's twiddle index is always 0,
// so W == 1: pure +-1 butterflies, real/imag independent).
// Pass 1: 64K-element contiguous chunk in 278KB LDS, 4 radix-16 groups (stages 0..15)
// Pass 2: 256x32 strided tile in LDS, 2 radix-16 groups (stages 16..23)
// Each radix-16 group = H16 * X via 4 accumulating V_WMMA_F32_16X16X4_F32.
// Global<->LDS traffic uses GLOBAL_LOAD_ASYNC_TO_LDS / STORE_ASYNC_FROM_LDS
// (ASYNCcnt) when the toolchain exposes the builtins.

typedef float v2f __attribute__((ext_vector_type(2)));
typedef float v8f __attribute__((ext_vector_type(8)));

#define THREADS 512
#define NWAVES  (THREADS / 32)
#define W2      41   // pass-2 LDS row pitch (odd; 2*41%64=18, 32*41%64=32 -> conflict-free)

#if __has_builtin(__builtin_amdgcn_global_load_async_to_lds_b32) && \
    __has_builtin(__builtin_amdgcn_global_store_async_from_lds_b32)
#define HAVE_ASYNC_LDS 1
typedef __attribute__((address_space(3))) int lds_int;   // AS3 ("__shared__") int*
typedef __attribute__((address_space(1))) int gbl_int;   // AS1 ("__device__") int*
#endif

__device__ __forceinline__ void wait_async0() {
#if __has_builtin(__builtin_amdgcn_s_wait_asynccnt)
  __builtin_amdgcn_s_wait_asynccnt(0);
#else
  asm volatile("s_wait_asynccnt 0" ::: "memory");
#endif
}

#ifdef HAVE_ASYNC_LDS
__device__ __forceinline__ void async_g2l(const float* g, float* l) {
  __builtin_amdgcn_global_load_async_to_lds_b32((gbl_int*)g, (lds_int*)l, 0, 0);
}
__device__ __forceinline__ void async_l2g(float* g, const float* l) {
  __builtin_amdgcn_global_store_async_from_lds_b32((gbl_int*)g, (lds_int*)l, 0, 0);
}
#endif

// Injective LDS pad map for pass 1: logical stride 16 -> physical 17 (conflict-free).
// Additive for 16-aligned deltas: pad16(a + d) == pad16(a) + pad16(d) when d%16==0.
__device__ __forceinline__ int pad16(int a) { return a + (a >> 4); }

// H16[m][k] = (-1)^popcount(m & k)
__device__ __forceinline__ float hsign(int m, int k) {
  return (__builtin_popcount(m & k) & 1) ? -1.0f : 1.0f;
}

// ---------------- Pass 1: stages 0..15 on contiguous 65536-element chunks ----
__global__ void __launch_bounds__(THREADS)
wht_low_kernel(const float* __restrict__ xr, const float* __restrict__ xi,
               float* __restrict__ out, int n) {
  extern __shared__ float ld[];                 // pad16(65536) = 69632 floats
  const int  arr  = blockIdx.y;
  const float* in = arr ? xi : xr;
  float*       op = out + (size_t)arr * (size_t)n;
  const size_t base = (size_t)blockIdx.x << 16;

  const int tid = threadIdx.x;
#ifdef HAVE_ASYNC_LDS
  for (int e = tid; e < 65536; e += THREADS)
    async_g2l(in + base + e, &ld[pad16(e)]);
  wait_async0();
#else
  for (int e = tid; e < 65536; e += THREADS)
    ld[pad16(e)] = in[base + e];
#endif
  __syncthreads();

  const int lane = tid & 31;
  const int wave = tid >> 5;
  const int half = lane >> 4;      // K half-select for A/B fragments
  const int col  = lane & 15;      // N (column) for B/C/D; M (row) for A

  // A-fragments: H16, constant per lane across all groups/tiles.
  v2f afrag[4];
#pragma unroll
  for (int kk = 0; kk < 4; ++kk) {
    const int k0 = 4 * kk + 2 * half;
    afrag[kk].x = hsign(col, k0);
    afrag[kk].y = hsign(col, k0 + 1);
  }

#pragma unroll
  for (int g = 0; g < 4; ++g) {                 // digit g, stride = 1 << (4g)
    const int sh = 4 * g;
    // Constant per-group LDS offsets (relative to tile base).
    int offL[8], offS[8];
#pragma unroll
    for (int kk = 0; kk < 4; ++kk) {
      const int r0 = 4 * kk + 2 * half;
      offL[2 * kk]     = pad16( r0      << sh);
      offL[2 * kk + 1] = pad16((r0 + 1) << sh);
    }
#pragma unroll
    for (int r = 0; r < 8; ++r)
      offS[r] = pad16((r + 8 * half) << sh);

    // First tile for this wave; constant physical stride between tiles.
    const int o0  = (wave << 4) | col;          // "other index" column
    const int ab0 = ((o0 >> sh) << (sh + 4)) | (o0 & ((1 << sh) - 1));
    int basep = pad16(ab0);
    const int dphys = (sh <= 8) ? (4096 + 256) : (256 + 16);

#pragma unroll 1
    for (int t = 0; t < 256 / NWAVES; ++t) {    // 16 disjoint tiles per wave
      v2f b[4];
#pragma unroll
      for (int kk = 0; kk < 4; ++kk) {
        b[kk].x = ld[basep + offL[2 * kk]];
        b[kk].y = ld[basep + offL[2 * kk + 1]];
      }
      v8f acc = {0.f, 0.f, 0.f, 0.f, 0.f, 0.f, 0.f, 0.f};
#pragma unroll
      for (int kk = 0; kk < 4; ++kk)
        acc = __builtin_amdgcn_wmma_f32_16x16x4_f32(
            false, afrag[kk], false, b[kk], (short)0, acc, false, false);
#pragma unroll
      for (int r = 0; r < 8; ++r)
        ld[basep + offS[r]] = acc[r];
      basep += dphys;
    }
    __syncthreads();
  }

#ifdef HAVE_ASYNC_LDS
  for (int e = tid; e < 65536; e += THREADS)
    async_l2g(op + base + e, &ld[pad16(e)]);
  // S_ENDPGM performs an implicit wait-idle (drains ASYNCcnt).
#else
  for (int e = tid; e < 65536; e += THREADS)
    op[base + e] = ld[pad16(e)];
#endif
}

// ---------------- Pass 2: stages 16..23, stride-65536 butterflies ------------
// Tile: 32 consecutive low indices x 256 strided high indices (m).
__global__ void __launch_bounds__(THREADS)
wht_high_kernel(float* __restrict__ data, int n) {
  extern __shared__ float ldt[];                // 256 * W2 floats
  const int arr = blockIdx.y;
  float* p = data + (size_t)arr * (size_t)n;
  const int lo0 = blockIdx.x << 5;

  const int tid = threadIdx.x;
#ifdef HAVE_ASYNC_LDS
  for (int e = tid; e < 8192; e += THREADS) {   // coalesced 128B rows
    const int m = e >> 5, j = e & 31;
    async_g2l(p + (size_t)m * 65536u + (size_t)(lo0 + j), &ldt[m * W2 + j]);
  }
  wait_async0();
#else
  for (int e = tid; e < 8192; e += THREADS) {
    const int m = e >> 5, j = e & 31;
    ldt[m * W2 + j] = p[(size_t)m * 65536u + (size_t)(lo0 + j)];
  }
#endif
  __syncthreads();

  const int lane = tid & 31;
  const int wave = tid >> 5;
  const int half = lane >> 4;
  const int col  = lane & 15;

  v2f afrag[4];
#pragma unroll
  for (int kk = 0; kk < 4; ++kk) {
    const int k0 = 4 * kk + 2 * half;
    afrag[kk].x = hsign(col, k0);
    afrag[kk].y = hsign(col, k0 + 1);
  }

  const int o0  = (wave << 4) | col;            // [0,512): 32 lo x 16 other-m
  const int j0  = o0 & 31;
  const int om0 = o0 >> 5;

#pragma unroll
  for (int g = 0; g < 2; ++g) {                 // g=0: low nibble of m, g=1: high
    int offL[8], offS[8], basep, dphys;
    if (g == 0) {                               // m = om*16 + row
#pragma unroll
      for (int kk = 0; kk < 4; ++kk) {
        const int r0 = 4 * kk + 2 * half;
        offL[2 * kk]     =  r0      * W2;
        offL[2 * kk + 1] = (r0 + 1) * W2;
      }
#pragma unroll
      for (int r = 0; r < 8; ++r) offS[r] = (r + 8 * half) * W2;
      basep = om0 * 16 * W2 + j0;
      dphys = 128 * W2;                         // om += 8 per tile
    } else {                                    // m = row*16 + om
#pragma unroll
      for (int kk = 0; kk < 4; ++kk) {
        const int r0 = 4 * kk + 2 * half;
        offL[2 * kk]     =  r0      * 16 * W2;
        offL[2 * kk + 1] = (r0 + 1) * 16 * W2;
      }
#pragma unroll
      for (int r = 0; r < 8; ++r) offS[r] = (r + 8 * half) * 16 * W2;
      basep = om0 * W2 + j0;
      dphys = 8 * W2;
    }

#pragma unroll 1
    for (int t = 0; t < 32 / NWAVES; ++t) {     // 2 disjoint tiles per wave
      v2f b[4];
#pragma unroll
      for (int kk = 0; kk < 4; ++kk) {
        b[kk].x = ldt[basep + offL[2 * kk]];
        b[kk].y = ldt[basep + offL[2 * kk + 1]];
      }
      v8f acc = {0.f, 0.f, 0.f, 0.f, 0.f, 0.f, 0.f, 0.f};
#pragma unroll
      for (int kk = 0; kk < 4; ++kk)
        acc = __builtin_amdgcn_wmma_f32_16x16x4_f32(
            false, afrag[kk], false, b[kk], (short)0, acc, false, false);
#pragma unroll
      for (int r = 0; r < 8; ++r)
        ldt[basep + offS[r]] = acc[r];
      basep += dphys;
    }
    __syncthreads();
  }

#ifdef HAVE_ASYNC_LDS
  for (int e = tid; e < 8192; e += THREADS) {
    const int m = e >> 5, j = e & 31;
    async_l2g(p + (size_t)m * 65536u + (size_t)(lo0 + j), &ldt[m * W2 + j]);
  }
#else
  for (int e = tid; e < 8192; e += THREADS) {
    const int m = e >> 5, j = e & 31;
    p[(size_t)m * 65536u + (size_t)(lo0 + j)] = ldt[m * W2 + j];
  }
#endif
}

extern "C" void kernel_launch(void* const* d_in, const int* in_sizes, int n_in,
                              void* d_out, int out_size, void* d_ws, size_t ws_size,
                              hipStream_t stream) {
  (void)n_in; (void)out_size; (void)d_ws; (void)ws_size;
  const float* xr = (const float*)d_in[0];
  const float* xi = (const float*)d_in[1];
  float* out = (float*)d_out;
  const int n = in_sizes[0];                    // 1 << 24

  const int chunks = n >> 16;                   // 256
  const size_t lds1 = (size_t)(65536 + 4096) * sizeof(float);  // 278528 B (<320KB WGP LDS)
  wht_low_kernel<<<dim3(chunks, 2), THREADS, lds1, stream>>>(xr, xi, out, n);

  const size_t lds2 = (size_t)256 * W2 * sizeof(float);
  wht_high_kernel<<<dim3(65536 / 32, 2), THREADS, lds2, stream>>>(out, n);
}